// RecurrentGCN_26963804684655
// MI455X (gfx1250) — compile-verified
//
#include <hip/hip_runtime.h>
#include <cmath>

typedef float v2f __attribute__((ext_vector_type(2)));
typedef float v8f __attribute__((ext_vector_type(8)));
typedef int   v4i_vs __attribute__((__vector_size__(16)));   // matches builtin param

#define N_NODES 262144
#define FDIM 128
#define HIDDIM 32
#define THREADS 256                    // 8 waves
#define ROWS_PER_TILE 128              // 8 waves x 16 rows
#define TILES_PER_BLOCK 4
#define LDS_STRIDE 132                 // 128 + 4 pad -> conflict-free b64 reads
#define TILE_LDS_FLOATS (ROWS_PER_TILE * LDS_STRIDE)   // 16896 floats
#define F4_PER_TILE (ROWS_PER_TILE * (FDIM / 4))       // 4096 float4
#define F4_PER_THREAD (F4_PER_TILE / THREADS)          // 16

#if defined(__has_builtin)
#if __has_builtin(__builtin_amdgcn_global_load_async_to_lds_b128)
#define HAVE_ASYNC_LDS 1
#endif
#if __has_builtin(__builtin_amdgcn_s_wait_asynccnt)
#define HAVE_WAIT_ASYNC 1
#endif
#endif

__device__ __forceinline__ void wait_async0() {
#ifdef HAVE_WAIT_ASYNC
  __builtin_amdgcn_s_wait_asynccnt(0);
#else
  asm volatile("s_wait_asynccnt 0x0" ::: "memory");
#endif
}

// One async 16-byte global -> LDS transfer per lane (ASYNCcnt-tracked).
__device__ __forceinline__ void async_copy16(const float* gp, float* lp) {
#ifdef HAVE_ASYNC_LDS
  __builtin_amdgcn_global_load_async_to_lds_b128(
      (__attribute__((address_space(1))) v4i_vs*)(uintptr_t)gp,
      (__attribute__((address_space(3))) v4i_vs*)(uint32_t)(uintptr_t)lp,
      0, 0);
#else
  asm volatile("global_load_async_to_lds_b128 %0, %1, off"
               :: "v"((uint32_t)(uintptr_t)lp), "v"(gp)
               : "memory");
#endif
}

// Stage one 128x128 f32 tile of x into padded LDS, asynchronously.
__device__ __forceinline__ void prefetch_tile(const float* __restrict__ gsrc,
                                              float* __restrict__ lbuf, int tid) {
#pragma unroll
  for (int i = 0; i < F4_PER_THREAD; ++i) {
    int flat = tid + i * THREADS;        // float4 units; 32 per row
    int row  = flat >> 5;
    int c4   = flat & 31;
    async_copy16(gsrc + (size_t)flat * 4, lbuf + row * LDS_STRIDE + c4 * 4);
  }
}

// Branch-free activations using v_exp_f32 + v_rcp_f32 (saturate via inf/0).
__device__ __forceinline__ float fast_sigmoid(float v) {
  return __builtin_amdgcn_rcpf(1.0f + __expf(-v));
}
__device__ __forceinline__ float fast_tanh(float v) {
  return 1.0f - 2.0f * __builtin_amdgcn_rcpf(__expf(2.0f * v) + 1.0f);
}

// ---------------------------------------------------------------------------
// Prep: swizzled effective weights (Weff = W[0]+W[1], rows 0..127) into d_ws,
// laid out in exact WMMA B-fragment lane order.
// slot = gate*4096 + ((s*2 + nhalf)*64 + 2*L + j)  ->  Weff[k][n],
//   k = 4*s + 2*(L>>4) + j,  n = 16*nhalf + (L&15)
// ---------------------------------------------------------------------------
__global__ void prep_weights(const float* __restrict__ Wz,
                             const float* __restrict__ Wh,
                             float* __restrict__ ws) {
  int tid = blockIdx.x * blockDim.x + threadIdx.x;   // 0..8191
  if (tid >= 8192) return;
  int gate = tid >> 12;
  int idx  = tid & 4095;
  int j  = idx & 1;
  int L  = (idx >> 1) & 31;
  int nh = (idx >> 6) & 1;
  int s  = idx >> 7;
  int k  = 4 * s + 2 * (L >> 4) + j;
  int n  = 16 * nh + (L & 15);
  const float* W = gate ? Wh : Wz;                   // shape (2,160,32) flat
  ws[tid] = W[k * HIDDIM + n] + W[160 * HIDDIM + k * HIDDIM + n];
}

// ---------------------------------------------------------------------------
// Fused DCRNN cell + readout; double-buffered async-LDS pipeline.
// ---------------------------------------------------------------------------
__global__ __launch_bounds__(THREADS) void dcrnn_fused(
    const float* __restrict__ x,      // [N,128]
    const float* __restrict__ wsw,    // swizzled weights [2][4096]
    const float* __restrict__ bz,     // [32]
    const float* __restrict__ bh,     // [32]
    const float* __restrict__ wlin,   // [64]
    const float* __restrict__ blin,   // [1]
    float* __restrict__ out)          // [N]
{
  extern __shared__ float lds[];      // 2 * TILE_LDS_FLOATS floats (dynamic)

  const int tid = threadIdx.x;
  const long long block_base =
      (long long)blockIdx.x * (ROWS_PER_TILE * TILES_PER_BLOCK);

  // Kick off tile 0 while we set up per-lane constants.
  prefetch_tile(x + block_base * FDIM, lds, tid);

  const int lane = tid & 31;
  const int wave = tid >> 5;           // 0..7
  const int wrow = wave * 16;
  const int half = lane >> 4;
  const int l15  = lane & 15;

  // Per-lane epilogue constants (column = l15 / 16+l15).
  const int c0 = l15, c1 = 16 + l15;
  const float bz0s = bz[c0], bz1s = bz[c1];
  const float bh0s = bh[c0], bh1s = bh[c1];
  const float w10 = wlin[c0],      w11 = wlin[c1];
  const float w20 = wlin[32 + c0], w21 = wlin[32 + c1];
  const float bl  = blin[0];

  const float* wz = wsw;               // gate z fragments
  const float* wh = wsw + 4096;        // gate h fragments

  for (int t = 0; t < TILES_PER_BLOCK; ++t) {
    wait_async0();                     // this wave's chunk of tile t landed
    __syncthreads();                   // everyone's chunk landed; buf (t+1)&1 free
    if (t + 1 < TILES_PER_BLOCK)
      prefetch_tile(x + (block_base + (long long)(t + 1) * ROWS_PER_TILE) * FDIM,
                    lds + ((t + 1) & 1) * TILE_LDS_FLOATS, tid);

    const float* buf = lds + (t & 1) * TILE_LDS_FLOATS;
    const float* lrow = &buf[(wrow + l15) * LDS_STRIDE + 2 * half];

    v8f accz0 = {}, accz1 = {}, acch0 = {}, acch1 = {};
#pragma unroll 4
    for (int s = 0; s < 32; ++s) {
      v2f a   = *(const v2f*)(lrow + 4 * s);                      // ds_load_b64
      v2f bz0 = *(const v2f*)(wz + (s * 2 + 0) * 64 + 2 * lane);  // 256B coalesced
      v2f bz1 = *(const v2f*)(wz + (s * 2 + 1) * 64 + 2 * lane);
      v2f bh0 = *(const v2f*)(wh + (s * 2 + 0) * 64 + 2 * lane);
      v2f bh1 = *(const v2f*)(wh + (s * 2 + 1) * 64 + 2 * lane);
      accz0 = __builtin_amdgcn_wmma_f32_16x16x4_f32(false, a, false, bz0, (short)0, accz0, false, false);
      accz1 = __builtin_amdgcn_wmma_f32_16x16x4_f32(false, a, false, bz1, (short)0, accz1, false, false);
      acch0 = __builtin_amdgcn_wmma_f32_16x16x4_f32(false, a, false, bh0, (short)0, acch0, false, false);
      acch1 = __builtin_amdgcn_wmma_f32_16x16x4_f32(false, a, false, bh1, (short)0, acch1, false, false);
    }

    // Epilogue. Reg g holds row g (lanes 0-15) / g+8 (lanes 16-31); lane = col.
#pragma unroll
    for (int g = 0; g < 8; ++g) {
      float z0  = fast_sigmoid(accz0[g] + bz0s);
      float z1  = fast_sigmoid(accz1[g] + bz1s);
      float t0  = fast_tanh(acch0[g] + bh0s);
      float t1  = fast_tanh(acch1[g] + bh1s);
      float h0v = (1.0f - z0) * t0;
      float h1v = (1.0f - z1) * t1;
      float c = fmaxf(h0v, 0.0f) * w10 + fast_sigmoid(h0v) * w20
              + fmaxf(h1v, 0.0f) * w11 + fast_sigmoid(h1v) * w21;
#pragma unroll
      for (int m = 8; m >= 1; m >>= 1)
        c += __shfl_xor(c, m, 32);
      if (l15 == 0) {
        long long node = block_base + (long long)t * ROWS_PER_TILE + wrow + g + 8 * half;
        out[node] = fmaxf(c + bl, 0.0f);
      }
    }
  }
}

// ---------------------------------------------------------------------------
extern "C" void kernel_launch(void* const* d_in, const int* in_sizes, int n_in,
                              void* d_out, int out_size, void* d_ws, size_t ws_size,
                              hipStream_t stream) {
  (void)in_sizes; (void)n_in; (void)out_size; (void)ws_size;
  // 0:x 1:edge_index 2:edge_weight 3:W_z 4:b_z 5:W_r 6:b_r 7:W_h 8:b_h 9:W_lin 10:b_lin
  const float* x    = (const float*)d_in[0];
  const float* Wz   = (const float*)d_in[3];
  const float* bzp  = (const float*)d_in[4];
  const float* Wh   = (const float*)d_in[7];
  const float* bhp  = (const float*)d_in[8];
  const float* Wlin = (const float*)d_in[9];
  const float* blin = (const float*)d_in[10];
  float* out = (float*)d_out;
  float* ws  = (float*)d_ws;

  prep_weights<<<32, 256, 0, stream>>>(Wz, Wh, ws);

  const int blocks = N_NODES / (ROWS_PER_TILE * TILES_PER_BLOCK);   // 512
  const size_t shmem = 2u * TILE_LDS_FLOATS * sizeof(float);        // ~132 KB
  dcrnn_fused<<<blocks, THREADS, shmem, stream>>>(
      x, ws, bzp, bhp, Wlin, blin, out);
}